// GCN_3573412790605
// MI455X (gfx1250) — compile-verified
//
#include <hip/hip_runtime.h>

// Problem sizes (fixed by the reference)
#define NN   50000
#define EE   600000
#define FDIM 128      // IN == H == 128
#define OUTD 10
#define GG   64

typedef __attribute__((ext_vector_type(2))) float v2f;
typedef __attribute__((ext_vector_type(8))) float v8f;

// ---------------------------------------------------------------------------
// Dense GEMM Y[N x 128] = X[N x 128] * W[128 x 128] using fp32 WMMA.
// Block = 256 threads = 8 wave32s; each wave computes one 16x16 tile;
// a block covers a 16-row x 128-col stripe. N % 16 == 0, so EXEC is all-1s.
// ---------------------------------------------------------------------------
__global__ __launch_bounds__(256) void gcn_gemm128_wmma(
    const float* __restrict__ X, const float* __restrict__ W,
    float* __restrict__ Y) {
  const int lane = threadIdx.x & 31;
  const int wave = threadIdx.x >> 5;
  const int m0   = blockIdx.x << 4;   // row tile
  const int n0   = wave << 4;         // col tile (8 waves * 16 = 128)
  const int r    = lane & 15;
  const int koff = (lane >> 4) << 1;  // lanes 16-31 hold K+2,K+3

  const float* __restrict__ xrow = X + (size_t)(m0 + r) * FDIM;

  v8f c = {};
#pragma unroll
  for (int k0 = 0; k0 < FDIM; k0 += 4) {
    v2f a, b;
    // A 16x4 fragment: V0 = {K=k0+koff}, V1 = {K=k0+koff+1}
    a.x = xrow[k0 + koff];
    a.y = xrow[k0 + koff + 1];
    // B 4x16 fragment: same K striping, N = n0 + r
    b.x = W[(size_t)(k0 + koff) * FDIM + n0 + r];
    b.y = W[(size_t)(k0 + koff + 1) * FDIM + n0 + r];
    c = __builtin_amdgcn_wmma_f32_16x16x4_f32(
        /*neg_a=*/false, a, /*neg_b=*/false, b,
        /*c_mod=*/(short)0, c, /*reuse_a=*/false, /*reuse_b=*/false);
  }

  // C/D 16x16: VGPR i -> row m0+i (lanes 0-15) / m0+i+8 (lanes 16-31)
  const int rowBase = m0 + ((lane >> 4) << 3);
#pragma unroll
  for (int i = 0; i < 8; ++i) {
    Y[(size_t)(rowBase + i) * FDIM + n0 + r] = c[i];
  }
}

// ---------------------------------------------------------------------------
// Degree / normalization helpers (dinv array doubles as the deg accumulator)
// ---------------------------------------------------------------------------
__global__ void deg_init(float* __restrict__ deg) {
  int i = blockIdx.x * blockDim.x + threadIdx.x;
  if (i < NN) deg[i] = 1.0f;  // +1 self loop
}

__global__ void deg_count(const int* __restrict__ dst, float* __restrict__ deg) {
  int e = blockIdx.x * blockDim.x + threadIdx.x;
  if (e < EE) atomicAdd(&deg[dst[e]], 1.0f);
}

__global__ void deg_to_rsqrt(float* __restrict__ deg) {
  int i = blockIdx.x * blockDim.x + threadIdx.x;
  if (i < NN) deg[i] = rsqrtf(deg[i]);
}

// ---------------------------------------------------------------------------
// out[i][f] = h[i][f] * (1/deg_i) + bias[f]     (self-loop term + bias)
// One thread per float4.
// ---------------------------------------------------------------------------
__global__ void self_bias_init(const float* __restrict__ h,
                               const float* __restrict__ dinv,
                               const float* __restrict__ bias,
                               float* __restrict__ out) {
  int t = blockIdx.x * blockDim.x + threadIdx.x;        // float4 index
  size_t idx = (size_t)t * 4;
  if (idx >= (size_t)NN * FDIM) return;
  int node = (int)(idx >> 7);                            // /128
  int f    = (int)(idx & 127);
  float di = dinv[node];
  float inv = di * di;                                   // 1/deg
  float4 hv = *(const float4*)(h + idx);
  float4 o;
  o.x = hv.x * inv + bias[f + 0];
  o.y = hv.y * inv + bias[f + 1];
  o.z = hv.z * inv + bias[f + 2];
  o.w = hv.w * inv + bias[f + 3];
  *(float4*)(out + idx) = o;
}

// ---------------------------------------------------------------------------
// Edge scatter: out[dst] += h[src] * dinv[src]*dinv[dst]
// 32 threads per edge, 4 feats per thread (coalesced 512B per edge row).
// ---------------------------------------------------------------------------
__global__ void edge_agg(const float* __restrict__ h,
                         const int* __restrict__ src,
                         const int* __restrict__ dst,
                         const float* __restrict__ dinv,
                         float* __restrict__ out) {
  long gid = (long)blockIdx.x * blockDim.x + threadIdx.x;
  long e = gid >> 5;
  if (e >= EE) return;
  int fg = (int)((gid & 31) << 2);
  int s = src[e], d = dst[e];
  float norm = dinv[s] * dinv[d];
  float4 hv = *(const float4*)(h + (size_t)s * FDIM + fg);
  float* o = out + (size_t)d * FDIM + fg;
  atomicAdd(o + 0, hv.x * norm);
  atomicAdd(o + 1, hv.y * norm);
  atomicAdd(o + 2, hv.z * norm);
  atomicAdd(o + 3, hv.w * norm);
}

__global__ void relu_inplace(float* __restrict__ p) {
  int t = blockIdx.x * blockDim.x + threadIdx.x;
  size_t idx = (size_t)t * 4;
  if (idx >= (size_t)NN * FDIM) return;
  float4 v = *(float4*)(p + idx);
  v.x = v.x > 0.f ? v.x : 0.f;
  v.y = v.y > 0.f ? v.y : 0.f;
  v.z = v.z > 0.f ? v.z : 0.f;
  v.w = v.w > 0.f ? v.w : 0.f;
  *(float4*)(p + idx) = v;
}

// ---------------------------------------------------------------------------
// Pooling
// ---------------------------------------------------------------------------
__global__ void zero_f(float* __restrict__ p, int n) {
  int i = blockIdx.x * blockDim.x + threadIdx.x;
  if (i < n) p[i] = 0.f;
}

__global__ void pool_sum(const float* __restrict__ h,
                         const int* __restrict__ batch,
                         float* __restrict__ pooled,
                         float* __restrict__ cnt) {
  long gid = (long)blockIdx.x * blockDim.x + threadIdx.x;
  long node = gid >> 5;
  if (node >= NN) return;
  int fg = (int)((gid & 31) << 2);
  int g = batch[node];
  float4 hv = *(const float4*)(h + (size_t)node * FDIM + fg);
  float* p = pooled + (size_t)g * FDIM + fg;
  atomicAdd(p + 0, hv.x);
  atomicAdd(p + 1, hv.y);
  atomicAdd(p + 2, hv.z);
  atomicAdd(p + 3, hv.w);
  if (fg == 0) atomicAdd(&cnt[g], 1.0f);
}

__global__ void final_fc(const float* __restrict__ pooled,
                         const float* __restrict__ cnt,
                         const float* __restrict__ Wfc,
                         const float* __restrict__ bfc,
                         float* __restrict__ out) {
  __shared__ float sh[FDIM];
  int g = blockIdx.x;
  int t = threadIdx.x;
  float c = cnt[g];
  c = c < 1.0f ? 1.0f : c;
  sh[t] = pooled[(size_t)g * FDIM + t] / c;
  __syncthreads();
  if (t < OUTD) {
    float acc = bfc[t];
    for (int f = 0; f < FDIM; ++f) acc += sh[f] * Wfc[f * OUTD + t];
    out[g * OUTD + t] = acc;
  }
}

// ---------------------------------------------------------------------------
// Host launcher
// ---------------------------------------------------------------------------
extern "C" void kernel_launch(void* const* d_in, const int* in_sizes, int n_in,
                              void* d_out, int out_size, void* d_ws, size_t ws_size,
                              hipStream_t stream) {
  const float* x    = (const float*)d_in[0];
  const int*   ei   = (const int*)d_in[1];
  const int*   src  = ei;
  const int*   dst  = ei + EE;
  const int*   batch= (const int*)d_in[2];
  const float* W1   = (const float*)d_in[3];
  const float* b1   = (const float*)d_in[4];
  const float* W2   = (const float*)d_in[5];
  const float* b2   = (const float*)d_in[6];
  const float* W3   = (const float*)d_in[7];
  const float* b3   = (const float*)d_in[8];
  const float* Wfc  = (const float*)d_in[9];
  const float* bfc  = (const float*)d_in[10];
  float* out = (float*)d_out;

  float* ws     = (float*)d_ws;
  float* bufA   = ws;                           // N*FDIM (GEMM result h)
  float* bufB   = bufA + (size_t)NN * FDIM;     // N*FDIM (aggregated features)
  float* dinv   = bufB + (size_t)NN * FDIM;     // N
  float* pooled = dinv + NN;                    // G*FDIM
  float* cnt    = pooled + (size_t)GG * FDIM;   // G

  const int TB = 256;
  const int nodeBlocks = (NN + TB - 1) / TB;
  const int edgeBlocks = (EE + TB - 1) / TB;
  const int elem4      = (NN * FDIM) / 4;                 // 1,600,000
  const int elem4Blocks= (elem4 + TB - 1) / TB;           // 6250
  const long edgeWork  = (long)EE * 32;                   // 19.2M
  const int edgeWorkBlocks = (int)((edgeWork + TB - 1) / TB);
  const long poolWork  = (long)NN * 32;
  const int poolBlocks = (int)((poolWork + TB - 1) / TB);

  // Degrees -> dinv = rsqrt(deg)  (recomputed every call; graph-safe)
  deg_init<<<nodeBlocks, TB, 0, stream>>>(dinv);
  deg_count<<<edgeBlocks, TB, 0, stream>>>(dst, dinv);
  deg_to_rsqrt<<<nodeBlocks, TB, 0, stream>>>(dinv);

  // 3 GCN layers. GEMM: feat -> bufA. Aggregate: bufA -> bufB. ReLU in place.
  const float* feat = x;
  const float* Ws[3] = {W1, W2, W3};
  const float* bs[3] = {b1, b2, b3};
  for (int l = 0; l < 3; ++l) {
    gcn_gemm128_wmma<<<NN / 16, TB, 0, stream>>>(feat, Ws[l], bufA);
    self_bias_init<<<elem4Blocks, TB, 0, stream>>>(bufA, dinv, bs[l], bufB);
    edge_agg<<<edgeWorkBlocks, TB, 0, stream>>>(bufA, src, dst, dinv, bufB);
    relu_inplace<<<elem4Blocks, TB, 0, stream>>>(bufB);
    feat = bufB;
  }

  // Global mean pool + FC
  zero_f<<<((GG * FDIM + GG) + TB - 1) / TB, TB, 0, stream>>>(pooled, GG * FDIM + GG);
  pool_sum<<<poolBlocks, TB, 0, stream>>>(bufB, batch, pooled, cnt);
  final_fc<<<GG, FDIM, 0, stream>>>(pooled, cnt, Wfc, bfc, out);
}